// PCENLayer_90477781058054
// MI455X (gfx1250) — compile-verified
//
#include <hip/hip_runtime.h>

#define FLOORV 1e-6f

constexpr int B = 64, C = 128, T = 4000;
constexpr int ROWS   = B * C;          // 8192 scan chains
constexpr int SEG    = 8;              // segments per chain (T split for parallelism)
constexpr int SEGLEN = T / SEG;        // 500
constexpr int TCHUNK = 100;            // elements staged per chunk per stream
constexpr int NCH    = SEGLEN / TCHUNK;    // 5 chunks
constexpr int F4S    = TCHUNK / 4;         // 25 float4 per stream per chunk
constexpr int LSTRIDE = 108;           // padded LDS row stride (floats, 16B aligned)

// ---- CDNA5 async global->LDS path (gated; fallback = synchronous copy) ----
#if defined(__HIP_DEVICE_COMPILE__) && \
    __has_builtin(__builtin_amdgcn_global_load_async_to_lds_b128) && \
    __has_builtin(__builtin_amdgcn_s_wait_asynccnt)
#define ASYNC_OK 1
typedef int v4i __attribute__((vector_size(16)));
typedef __attribute__((address_space(1))) v4i gv4i;   // global int4
typedef __attribute__((address_space(3))) v4i lv4i;   // LDS int4
#define ASYNC_WAIT(n) __builtin_amdgcn_s_wait_asynccnt(n)
#else
#define ASYNC_OK 0
#define ASYNC_WAIT(n) ((void)0)
#endif

__device__ __forceinline__ void stage_f4(const float* g, float* l) {
#if ASYNC_OK
  __builtin_amdgcn_global_load_async_to_lds_b128(
      (gv4i*)g, (lv4i*)l, /*offset=*/0, /*cpol=*/0);
#else
  *(float4*)l = *(const float4*)g;
#endif
}

// ---------------------------------------------------------------------------
// Pass 1: per-(row, segment) local EMA state with zero incoming state
// (segment 0 carries the true init = x[...,0]). Linear recurrence =>
// true state folds later as S_s = (1-w)^500 * S_{s-1} + local_s.
// bnd laid out with stride 8 (slot 7 unused) so row/seg are shift/mask.
// ---------------------------------------------------------------------------
__global__ void __launch_bounds__(256)
pcen_boundary(const float* __restrict__ x, const float* __restrict__ ema_w,
              float* __restrict__ bnd) {
  int u = blockIdx.x * blockDim.x + threadIdx.x;   // u = row*8 + s
  int s = u & (SEG - 1);
  if (s == SEG - 1) return;                        // segment 7 boundary never needed
  int row = u >> 3;
  int c   = row & (C - 1);

  float w   = fminf(fmaxf(ema_w[c], 0.02f), 1.0f);
  float omw = 1.0f - w;

  const float* p = x + (size_t)row * T + (size_t)s * SEGLEN;
  float acc = (s == 0) ? p[0] : 0.0f;          // init = x[...,0] for segment 0
  const float4* p4 = (const float4*)p;         // 2000B segment offset -> 16B aligned
#pragma unroll 1
  for (int i = 0; i < SEGLEN / 4; ++i) {
    float4 v = p4[i];
    acc = fmaf(w, v.x, omw * acc);
    acc = fmaf(w, v.y, omw * acc);
    acc = fmaf(w, v.z, omw * acc);
    acc = fmaf(w, v.w, omw * acc);
  }
  bnd[u] = acc;
}

// ---------------------------------------------------------------------------
// Pass 2: one thread per (row, segment). Double-buffered async staging of the
// wave's 32 contiguous streams into LDS, then sequential scan + pointwise.
// Each wave stages only its own lanes' data -> per-wave ASYNCcnt sync, no barriers.
// ---------------------------------------------------------------------------
__global__ void __launch_bounds__(64)
pcen_main(const float* __restrict__ x,
          const float* __restrict__ alpha, const float* __restrict__ delta,
          const float* __restrict__ root,  const float* __restrict__ ema_w,
          const float* __restrict__ bnd,   float* __restrict__ out) {
  __shared__ float tile[2][2][32][LSTRIDE];    // [wave][buf][stream][padded chunk]

  const int tid  = threadIdx.x;
  const int wave = tid >> 5, lane = tid & 31;
  const int unit = blockIdx.x * 64 + tid;      // unit = row*SEG + s, contiguous in memory
  const int row  = unit >> 3;
  const int s    = unit & (SEG - 1);
  const int c    = row & (C - 1);
  const int wunit0 = blockIdx.x * 64 + wave * 32;

  // per-channel parameters
  float w     = fminf(fmaxf(ema_w[c], 0.02f), 1.0f);
  float omw   = 1.0f - w;
  float a     = fminf(alpha[c], 1.0f);
  float na    = -a;
  float inv_r = 1.0f / fmaxf(root[c], 1.0f);
  float d     = delta[c];
  float dpow  = exp2f(inv_r * __log2f(d));
  float Apow  = exp2f((float)SEGLEN * __log2f(omw));   // (1-w)^500 (omw=0 -> 0)

  // incoming EMA state for this segment
  float acc;
  if (s == 0) {
    acc = x[(size_t)row * T];                  // init; first update reproduces x0
  } else {
    const float* b = bnd + (size_t)row * SEG;  // stride-8 boundary slots
    float S = b[0];                            // S_0
    for (int j = 1; j < s; ++j) S = fmaf(Apow, S, b[j]);
    acc = S;                                   // S_{s-1}
  }

  auto stage_chunk = [&](int ci, int bb) {
    if (lane < F4S) {
      const float* gbase = x + (size_t)wunit0 * SEGLEN + (size_t)ci * TCHUNK
                             + (size_t)lane * 4;
      float* lbase = &tile[wave][bb][0][lane * 4];
#pragma unroll
      for (int st = 0; st < 32; ++st)          // one coalesced 400B request per stream
        stage_f4(gbase + (size_t)st * SEGLEN, lbase + st * LSTRIDE);
    }
  };

  stage_chunk(0, 0);

  const size_t base_elem = (size_t)unit * SEGLEN;
  float4 o4;
  for (int ci = 0; ci < NCH; ++ci) {
    const int bb = ci & 1;
    if (ci + 1 < NCH) {
      stage_chunk(ci + 1, bb ^ 1);             // prefetch next chunk
      ASYNC_WAIT(32);                          // in-order loads: current chunk done
    } else {
      ASYNC_WAIT(0);
    }

    const float* lsrc = &tile[wave][bb][lane][0];
    float* obase = out + base_elem + (size_t)ci * TCHUNK;
#pragma unroll 4
    for (int k = 0; k < TCHUNK; ++k) {
      float xv = lsrc[k];
      acc = fmaf(w, xv, omw * acc);                       // EMA step
      float pinv = exp2f(na * __log2f(FLOORV + acc));     // (eps+ema)^-a
      float u    = fmaf(xv, pinv, d);                     // x*(..)^-a + d
      float o    = exp2f(inv_r * __log2f(u)) - dpow;      // (..)^(1/r) - d^(1/r)
      (&o4.x)[k & 3] = o;
      if ((k & 3) == 3) *(float4*)(obase + (k - 3)) = o4; // 16B stores, lane-contiguous
    }
  }
}

extern "C" void kernel_launch(void* const* d_in, const int* in_sizes, int n_in,
                              void* d_out, int out_size, void* d_ws, size_t ws_size,
                              hipStream_t stream) {
  const float* x     = (const float*)d_in[0];
  const float* alpha = (const float*)d_in[1];
  const float* delta = (const float*)d_in[2];
  const float* root  = (const float*)d_in[3];
  const float* emaw  = (const float*)d_in[4];
  float* out = (float*)d_out;
  float* bnd = (float*)d_ws;                   // ROWS*SEG floats = 256 KiB scratch

  pcen_boundary<<<(ROWS * SEG + 255) / 256, 256, 0, stream>>>(x, emaw, bnd);
  pcen_main<<<(ROWS * SEG) / 64, 64, 0, stream>>>(x, alpha, delta, root, emaw, bnd, out);
}